// Memory_40535901340444
// MI455X (gfx1250) — compile-verified
//
#include <hip/hip_runtime.h>
#include <math.h>

typedef __attribute__((ext_vector_type(2))) float v2f;
typedef __attribute__((ext_vector_type(4))) float v4f;
typedef __attribute__((ext_vector_type(8))) float v8f;

#define B_ 32
#define N_ 1024
#define W_ 128
#define R_ 4
#define EPS_ 1e-8f

__device__ inline v2f ld2(const float* p) { return *(const v2f*)p; }

__device__ inline v8f wmma4(v2f a, v2f b, v8f c) {
    return __builtin_amdgcn_wmma_f32_16x16x4_f32(
        /*neg_a=*/false, a, /*neg_b=*/false, b,
        /*c_mod=*/(short)0, c, /*reuse_a=*/false, /*reuse_b=*/false);
}

// ---------------------------------------------------------------------------
// Kernel 0: zero the global sum(ww) accumulator
// ---------------------------------------------------------------------------
__global__ void init_kernel(float* sum_ww) { *sum_ww = 0.f; }

// ---------------------------------------------------------------------------
// Kernel 1: cosine dot products for 4 read keys + 1 write key vs memory rows.
// One wave (32 lanes) per (batch, 16-column N-tile). A = keys (rows 0-3 read,
// row 4 write, rest zero-masked), B = memory rows, K-loop over W in steps of 4.
// Also accumulates key / memory-row squared norms in the same pass.
// Output: raw cosine cosAll[b, row(0..4), n].
// ---------------------------------------------------------------------------
__global__ __launch_bounds__(32)
void cos_kernel(const float* __restrict__ memory,
                const float* __restrict__ read_keys,
                const float* __restrict__ write_key,
                float* __restrict__ cosAll) {
    int b    = blockIdx.x >> 6;
    int nb   = (blockIdx.x & 63) * 16;
    int lane = threadIdx.x;
    int half = lane >> 4;
    int lr   = lane & 15;

    const float* kptr; float kmask = 1.f;
    if (lr < R_)       kptr = read_keys + (size_t)(b * R_ + lr) * W_;
    else if (lr == 4)  kptr = write_key + (size_t)b * W_;
    else             { kptr = read_keys + (size_t)b * R_ * W_; kmask = 0.f; }

    const float* mptr = memory + ((size_t)b * N_ + nb + lr) * W_;

    v8f acc = {};
    float kss = 0.f, mss = 0.f;
    #pragma unroll 4
    for (int s = 0; s < W_ / 4; ++s) {
        int w0 = 4 * s + 2 * half;
        v2f av = ld2(kptr + w0); av *= kmask;
        v2f bv = ld2(mptr + w0);
        kss += av.x * av.x + av.y * av.y;
        mss += bv.x * bv.x + bv.y * bv.y;
        acc = wmma4(av, bv, acc);
    }
    kss += __shfl_xor(kss, 16, 32);
    mss += __shfl_xor(mss, 16, 32);

    __shared__ float sKn[16], sMn[16];
    if (lane < 16) { sKn[lane] = sqrtf(kss); sMn[lane] = sqrtf(mss); }
    __syncthreads();

    #pragma unroll
    for (int v = 0; v < 8; ++v) {
        int M = v + 8 * half;
        if (M < 5) {
            float c = acc[v] / fmaxf(sKn[M] * sMn[lr], EPS_);
            cosAll[(size_t)(b * 5 + M) * N_ + nb + lr] = c;
        }
    }
}

// ---------------------------------------------------------------------------
// Kernel 2: softmax over N per (b,row). Rows 0-3: softmax(raw)*read_strength
// -> content_r. Row 4: softmax(raw*write_strength) -> wc.
// ---------------------------------------------------------------------------
__global__ __launch_bounds__(256)
void softmax_kernel(const float* __restrict__ cosAll,
                    const float* __restrict__ read_strength,
                    const float* __restrict__ write_strength,
                    float* __restrict__ content_r,
                    float* __restrict__ wc) {
    int b = blockIdx.x / 5;
    int row = blockIdx.x % 5;
    int t = threadIdx.x;
    const float* x = cosAll + (size_t)(b * 5 + row) * N_;
    float scale = (row == 4) ? write_strength[b] : 1.f;

    float v[4];
    float m = -3.402823466e+38f;
    #pragma unroll
    for (int j = 0; j < 4; ++j) { v[j] = x[t + j * 256] * scale; m = fmaxf(m, v[j]); }

    __shared__ float red[256];
    red[t] = m; __syncthreads();
    for (int off = 128; off > 0; off >>= 1) {
        if (t < off) red[t] = fmaxf(red[t], red[t + off]);
        __syncthreads();
    }
    m = red[0]; __syncthreads();

    float s = 0.f;
    #pragma unroll
    for (int j = 0; j < 4; ++j) { v[j] = __expf(v[j] - m); s += v[j]; }
    red[t] = s; __syncthreads();
    for (int off = 128; off > 0; off >>= 1) {
        if (t < off) red[t] += red[t + off];
        __syncthreads();
    }
    float inv = 1.f / red[0];

    if (row < 4) {
        float rs = read_strength[b * R_ + row];
        #pragma unroll
        for (int j = 0; j < 4; ++j)
            content_r[(size_t)(b * R_ + row) * N_ + t + j * 256] = v[j] * inv * rs;
    } else {
        #pragma unroll
        for (int j = 0; j < 4; ++j)
            wc[(size_t)b * N_ + t + j * 256] = v[j] * inv;
    }
}

// ---------------------------------------------------------------------------
// Kernel 3: allocation (exclusive cumprod of usage over N), write weights ww,
// and global scalar sum(ww) via atomic. One block of 256 per batch, 4 elems
// each, Hillis-Steele product scan across threads.
// ---------------------------------------------------------------------------
__global__ __launch_bounds__(256)
void ww_kernel(const float* __restrict__ usage,
               const float* __restrict__ wc,
               const float* __restrict__ allocation_gate,
               const float* __restrict__ write_gate,
               float* __restrict__ ww,
               float* __restrict__ sum_ww) {
    int b = blockIdx.x;
    int t = threadIdx.x;
    const float* u = usage + (size_t)b * N_;
    float uu[4];
    #pragma unroll
    for (int j = 0; j < 4; ++j) uu[j] = u[4 * t + j];
    float p = uu[0] * uu[1] * uu[2] * uu[3];

    __shared__ float sp[256];
    sp[t] = p; __syncthreads();
    for (int off = 1; off < 256; off <<= 1) {
        float val = sp[t];
        if (t >= off) val *= sp[t - off];
        __syncthreads();
        sp[t] = val;
        __syncthreads();
    }
    float excl = (t == 0) ? 1.f : sp[t - 1];

    float ag = allocation_gate[b], wg = write_gate[b];
    const float* wcb = wc + (size_t)b * N_;
    float lsum = 0.f;
    #pragma unroll
    for (int j = 0; j < 4; ++j) {
        float alloc = (1.f - uu[j]) * excl;
        float c = wcb[4 * t + j];
        float w = (ag * (alloc - c) + c) * wg;
        ww[(size_t)b * N_ + 4 * t + j] = w;
        lsum += w;
        excl *= uu[j];
    }
    __syncthreads();
    sp[t] = lsum; __syncthreads();
    for (int off = 128; off > 0; off >>= 1) {
        if (t < off) sp[t] += sp[t + off];
        __syncthreads();
    }
    if (t == 0) atomicAdd(sum_ww, sp[0]);
}

// ---------------------------------------------------------------------------
// Kernel 4: fw = rw @ L^T, bw = rw @ L via WMMA, fused with the mode mix into
// new_rw. One wave per (batch, 16-col m-tile); K-loop of 256 steps over N.
// L is re-read by link_kernel afterwards, so keep its lines RT (L2-resident:
// 134 MB < 192 MB L2).
// ---------------------------------------------------------------------------
__global__ __launch_bounds__(32)
void fwbw_kernel(const float* __restrict__ L,
                 const float* __restrict__ read_weights,
                 const float* __restrict__ read_modes,
                 const float* __restrict__ content_r,
                 float* __restrict__ new_rw_out) {
    int b    = blockIdx.x >> 6;
    int mb   = (blockIdx.x & 63) * 16;
    int lane = threadIdx.x;
    int half = lane >> 4;
    int lr   = lane & 15;

    float amask = (lr < R_) ? 1.f : 0.f;
    const float* aptr = read_weights + (size_t)(b * R_ + (lr < R_ ? lr : 0)) * N_;
    const float* Lb   = L + (size_t)b * N_ * N_;
    const float* frow = Lb + (size_t)(mb + lr) * N_;   // fw reads rows of L
    const float* bcol = Lb + (mb + lr);                // bw reads columns of L

    v8f accF = {}, accB = {};
    for (int s = 0; s < N_ / 4; ++s) {
        int k0 = 4 * s + 2 * half;
        v2f av = ld2(aptr + k0); av *= amask;
        v2f fb = ld2(frow + k0);
        v2f bb;
        bb.x = bcol[(size_t)k0 * N_];
        bb.y = bcol[(size_t)(k0 + 1) * N_];
        accF = wmma4(av, fb, accF);
        accB = wmma4(av, bb, accB);
    }

    if (half == 0) {
        #pragma unroll
        for (int r = 0; r < R_; ++r) {
            float m0 = read_modes[(b * R_ + r) * 3 + 0];
            float m1 = read_modes[(b * R_ + r) * 3 + 1];
            float m2 = read_modes[(b * R_ + r) * 3 + 2];
            float cr = content_r[(size_t)(b * R_ + r) * N_ + mb + lr];
            new_rw_out[(size_t)(b * R_ + r) * N_ + mb + lr] =
                accF[r] * m0 + accB[r] * m1 + cr * m2;
        }
    }
}

// ---------------------------------------------------------------------------
// Kernel 5: read_result = new_rw @ memory via WMMA. One wave per
// (batch, 16-col w-tile); K-loop of 256 steps over N.
// ---------------------------------------------------------------------------
__global__ __launch_bounds__(32)
void rr_kernel(const float* __restrict__ memory,
               const float* __restrict__ new_rw,
               float* __restrict__ read_result) {
    int b    = blockIdx.x >> 3;
    int wb   = (blockIdx.x & 7) * 16;
    int lane = threadIdx.x;
    int half = lane >> 4;
    int lr   = lane & 15;

    float amask = (lr < R_) ? 1.f : 0.f;
    const float* aptr = new_rw + (size_t)(b * R_ + (lr < R_ ? lr : 0)) * N_;
    const float* mcol = memory + (size_t)b * N_ * W_ + wb + lr;

    v8f acc = {};
    for (int s = 0; s < N_ / 4; ++s) {
        int k0 = 4 * s + 2 * half;
        v2f av = ld2(aptr + k0); av *= amask;
        v2f bv;
        bv.x = mcol[(size_t)k0 * W_];
        bv.y = mcol[(size_t)(k0 + 1) * W_];
        acc = wmma4(av, bv, acc);
    }
    if (half == 0) {
        #pragma unroll
        for (int r = 0; r < R_; ++r)
            read_result[(size_t)(b * R_ + r) * W_ + wb + lr] = acc[r];
    }
}

// ---------------------------------------------------------------------------
// Kernel 6: memory erase/write (elementwise, 128-bit streaming). This is the
// LAST reader of `memory` and `mem_out` is never re-read -> NT load + NT
// store to avoid occupying L2 lines needed for temporal_link.
// ---------------------------------------------------------------------------
__global__ __launch_bounds__(256)
void mem_kernel(const float* __restrict__ memory,
                const float* __restrict__ erase_vector,
                const float* __restrict__ write_vector,
                const float* __restrict__ ww,
                float* __restrict__ mem_out) {
    int idx = blockIdx.x * 256 + threadIdx.x;   // B*N*W/4 = 1048576 v4fs
    int wq = idx & 31;                          // W/4
    int n  = (idx >> 5) & 1023;
    int b  = idx >> 15;
    float w = ww[b * N_ + n];
    v4f m = __builtin_nontemporal_load((const v4f*)memory + idx);
    v4f e = ((const v4f*)erase_vector)[b * 32 + wq];
    v4f v = ((const v4f*)write_vector)[b * 32 + wq];
    v4f o;
    o.x = m.x * (1.f - w * e.x) + w * v.x;
    o.y = m.y * (1.f - w * e.y) + w * v.y;
    o.z = m.z * (1.f - w * e.z) + w * v.z;
    o.w = m.w * (1.f - w * e.w) + w * v.w;
    __builtin_nontemporal_store(o, (v4f*)mem_out + idx);
}

// ---------------------------------------------------------------------------
// Kernel 7: usage & precedence update (elementwise over B*N)
// ---------------------------------------------------------------------------
__global__ __launch_bounds__(256)
void up_kernel(const float* __restrict__ usage,
               const float* __restrict__ precedence,
               const float* __restrict__ ww,
               const float* __restrict__ free_gates,
               const float* __restrict__ new_rw,
               const float* __restrict__ sum_ww,
               float* __restrict__ usage_out,
               float* __restrict__ prec_out) {
    int i = blockIdx.x * 256 + threadIdx.x;     // B*N = 32768
    int b = i >> 10;
    int n = i & 1023;
    float w = ww[i];
    float u = usage[i];
    float ret = 1.f;
    #pragma unroll
    for (int r = 0; r < R_; ++r)
        ret *= (1.f - free_gates[b * R_ + r] * new_rw[(size_t)(b * R_ + r) * N_ + n]);
    usage_out[i] = (u + w - u * w) * ret;
    prec_out[i] = (1.f - sum_ww[0]) * precedence[i] + w;
}

// ---------------------------------------------------------------------------
// Kernel 8: temporal link update (the big 268 MB streaming kernel, 128-bit)
// link_new[b,i,j] = (1 - ww_i - ww_j) * L[b,i,j] + ww_i * prec_j
// Last use of L -> NT load (hit L2 lines left by fwbw, then drop); link_out
// never re-read -> NT store (don't evict anything useful).
// ---------------------------------------------------------------------------
__global__ __launch_bounds__(256)
void link_kernel(const float* __restrict__ L,
                 const float* __restrict__ ww,
                 const float* __restrict__ precedence,
                 float* __restrict__ link_out) {
    int idx = blockIdx.x * 256 + threadIdx.x;   // B*N*N/4 = 8388608 v4fs
    int jq = idx & 255;                         // N/4
    int i  = (idx >> 8) & 1023;
    int b  = idx >> 18;
    float wi = ww[b * N_ + i];
    v4f wj = ((const v4f*)(ww + (size_t)b * N_))[jq];
    v4f pj = ((const v4f*)(precedence + (size_t)b * N_))[jq];
    v4f l = __builtin_nontemporal_load((const v4f*)L + idx);
    v4f o;
    o.x = (1.f - wi - wj.x) * l.x + wi * pj.x;
    o.y = (1.f - wi - wj.y) * l.y + wi * pj.y;
    o.z = (1.f - wi - wj.z) * l.z + wi * pj.z;
    o.w = (1.f - wi - wj.w) * l.w + wi * pj.w;
    __builtin_nontemporal_store(o, (v4f*)link_out + idx);
}

// ---------------------------------------------------------------------------
extern "C" void kernel_launch(void* const* d_in, const int* in_sizes, int n_in,
                              void* d_out, int out_size, void* d_ws, size_t ws_size,
                              hipStream_t stream) {
    const float* memory          = (const float*)d_in[0];
    const float* temporal_link   = (const float*)d_in[1];
    const float* usage           = (const float*)d_in[2];
    const float* precedence      = (const float*)d_in[3];
    const float* read_weights    = (const float*)d_in[4];
    const float* read_keys       = (const float*)d_in[5];
    const float* read_strength   = (const float*)d_in[6];
    const float* read_modes      = (const float*)d_in[7];
    const float* write_key       = (const float*)d_in[8];
    const float* write_strength  = (const float*)d_in[9];
    const float* allocation_gate = (const float*)d_in[10];
    const float* write_gate      = (const float*)d_in[11];
    const float* write_vector    = (const float*)d_in[12];
    const float* erase_vector    = (const float*)d_in[13];
    const float* free_gates      = (const float*)d_in[14];

    float* out       = (float*)d_out;
    float* rr_out    = out;                          // (B,R,W)   16384
    float* mem_out   = out + 16384;                  // (B,N,W)   4194304
    float* link_out  = out + 4210688;                // (B,N,N)   33554432
    float* usage_out = out + 37765120;               // (B,1,N)   32768
    float* prec_out  = out + 37797888;               // (B,1,N)   32768
    float* nrw_out   = out + 37830656;               // (B,R,N)   131072

    float* ws        = (float*)d_ws;
    float* cosAll    = ws;                           // B*5*N  = 163840
    float* content_r = ws + 163840;                  // B*R*N  = 131072
    float* wc        = ws + 294912;                  // B*N    = 32768
    float* ww        = ws + 327680;                  // B*N    = 32768
    float* sum_ww    = ws + 360448;                  // 1

    init_kernel<<<1, 1, 0, stream>>>(sum_ww);
    cos_kernel<<<B_ * 64, 32, 0, stream>>>(memory, read_keys, write_key, cosAll);
    softmax_kernel<<<B_ * 5, 256, 0, stream>>>(cosAll, read_strength, write_strength,
                                               content_r, wc);
    ww_kernel<<<B_, 256, 0, stream>>>(usage, wc, allocation_gate, write_gate, ww, sum_ww);
    fwbw_kernel<<<B_ * 64, 32, 0, stream>>>(temporal_link, read_weights, read_modes,
                                            content_r, nrw_out);
    rr_kernel<<<B_ * 8, 32, 0, stream>>>(memory, nrw_out, rr_out);
    mem_kernel<<<4096, 256, 0, stream>>>(memory, erase_vector, write_vector, ww, mem_out);
    up_kernel<<<128, 256, 0, stream>>>(usage, precedence, ww, free_gates, nrw_out,
                                       sum_ww, usage_out, prec_out);
    link_kernel<<<32768, 256, 0, stream>>>(temporal_link, ww, precedence, link_out);
}